// TrainableCloud_69544110457119
// MI455X (gfx1250) — compile-verified
//
#include <hip/hip_runtime.h>
#include <math.h>

#define N_RAYS      65536
#define N_SAMPLES   256
#define SIGMA_SCALE 50.0f
#define GRID_N      256
#define GRID_ELEMS  (GRID_N * GRID_N * GRID_N)
#define REPACK_BYTES ((size_t)GRID_ELEMS * sizeof(float2))

__device__ __forceinline__ float clamp01(float v) {
    return fminf(fmaxf(v, 0.0f), 1.0f);   // lowers to v_max_num_f32 ... clamp
}

// ---------------------------------------------------------------------------
// Pass 1: repack grid -> z-paired, pre-clamped float2 layout (134 MB, fully
// L2-resident on MI455X's 192 MB L2). Streaming, HBM-bound (~8 us).
//   g2[x][y][z] = ( clamp01(g[x][y][z]), clamp01(g[x][y][min(z+1,255)]) )
// ---------------------------------------------------------------------------
__global__ __launch_bounds__(256)
void repack_kernel(const float* __restrict__ g, float2* __restrict__ g2)
{
    const int idx = blockIdx.x * 256 + threadIdx.x;   // 16.7M threads
    const int z   = idx & (GRID_N - 1);
    const float a = g[idx];
    const float b = g[idx + ((z < GRID_N - 1) ? 1 : 0)];
    float2 v;
    v.x = clamp01(a);
    v.y = clamp01(b);
    g2[idx] = v;
}

// ---------------------------------------------------------------------------
// Pass 2: one thread per ray. Explicit group-of-4 software pipeline:
//   phase 1: 4 independent address chains (fmaf(k,dg,g), no serial dep)
//   phase 2: 16 b64 gathers issued back-to-back (one clause, max MLP)
//   phase 3: trilerp + accumulate under progressive s_wait_loadcnt
// Transmittance telescopes: opacity = 1 - exp(-SIGMA*dt*sum(trilerp)).
// ---------------------------------------------------------------------------
template <bool PACKED>
__global__ __launch_bounds__(256)
void nerf_opacity_kernel(const float*  __restrict__ rays,
                         const float*  __restrict__ grid,
                         const float2* __restrict__ grid2,
                         float*        __restrict__ out)
{
    const int ray = blockIdx.x * blockDim.x + threadIdx.x;
    if (ray >= N_RAYS) return;

    // rays row = 6 floats, 24B stride -> 8B aligned: three float2 loads
    const float2* r2 = reinterpret_cast<const float2*>(rays) + ray * 3;
    const float2 p01 = r2[0];   // ox, oy
    const float2 p23 = r2[1];   // oz, dx
    const float2 p45 = r2[2];   // dy, dz
    const float ox = p01.x, oy = p01.y, oz = p23.x;
    float dx = p23.y, dy = p45.x, dz = p45.y;

    // normalize direction
    const float inv_len = rsqrtf(dx * dx + dy * dy + dz * dz);
    dx *= inv_len; dy *= inv_len; dz *= inv_len;

    // safe reciprocals
    const float ivx = 1.0f / ((fabsf(dx) < 1e-9f) ? 1e-9f : dx);
    const float ivy = 1.0f / ((fabsf(dy) < 1e-9f) ? 1e-9f : dy);
    const float ivz = 1.0f / ((fabsf(dz) < 1e-9f) ? 1e-9f : dz);

    // slab test against [-1,1]^3
    const float t0x = (-1.0f - ox) * ivx, t1x = (1.0f - ox) * ivx;
    const float t0y = (-1.0f - oy) * ivy, t1y = (1.0f - oy) * ivy;
    const float t0z = (-1.0f - oz) * ivz, t1z = (1.0f - oz) * ivz;
    const float tn = fmaxf(fmaxf(fmaxf(fminf(t0x, t1x), fminf(t0y, t1y)),
                                 fminf(t0z, t1z)), 0.0f);
    const float tf = fminf(fminf(fmaxf(t0x, t1x), fmaxf(t0y, t1y)),
                           fmaxf(t0z, t1z));

    if (!(tf > tn)) { out[ray] = 0.0f; return; }

    const float dt = (tf - tn) * (1.0f / (float)N_SAMPLES);

    // march in grid space: g = (p + 1) * 127.5
    const float t_start = tn + 0.5f * dt;
    float gx = fmaf(fmaf(t_start, dx, ox), 127.5f, 127.5f);
    float gy = fmaf(fmaf(t_start, dy, oy), 127.5f, 127.5f);
    float gz = fmaf(fmaf(t_start, dz, oz), 127.5f, 127.5f);
    const float dgx = dt * dx * 127.5f;
    const float dgy = dt * dy * 127.5f;
    const float dgz = dt * dz * 127.5f;
    const float dgx4 = 4.0f * dgx, dgy4 = 4.0f * dgy, dgz4 = 4.0f * dgz;

    float acc = 0.0f;

    for (int i = 0; i < N_SAMPLES; i += 4) {
        // CDNA5 global_prefetch_b8: warm both x-planes 16 samples ahead
        {
            float pfx = fminf(fmaxf(fmaf(16.0f, dgx, gx), 0.0f), 255.0f);
            float pfy = fminf(fmaxf(fmaf(16.0f, dgy, gy), 0.0f), 255.0f);
            float pfz = fminf(fmaxf(fmaf(16.0f, dgz, gz), 0.0f), 255.0f);
            int pix = min((int)pfx, 254);
            int piy = min((int)pfy, 254);
            int piz = min((int)pfz, 254);
            int pbase = (pix << 16) + (piy << 8) + piz;
            if (PACKED) {
                __builtin_prefetch(grid2 + pbase, 0, 0);
                __builtin_prefetch(grid2 + pbase + 65536, 0, 0);
            } else {
                __builtin_prefetch(grid + pbase, 0, 0);
                __builtin_prefetch(grid + pbase + 65536, 0, 0);
            }
        }

        // ---- phase 1: four independent address chains -------------------
        int   base_[4];
        float fx_[4], fy_[4], fz_[4];
        #pragma unroll
        for (int k = 0; k < 4; ++k) {
            const float kf = (float)k;
            const float cx = fminf(fmaxf(fmaf(kf, dgx, gx), 0.0f), 255.0f);
            const float cy = fminf(fmaxf(fmaf(kf, dgy, gy), 0.0f), 255.0f);
            const float cz = fminf(fmaxf(fmaf(kf, dgz, gz), 0.0f), 255.0f);
            const int ix = min((int)cx, 254);   // coords >= 0: trunc == floor
            const int iy = min((int)cy, 254);
            const int iz = min((int)cz, 254);
            fx_[k] = cx - (float)ix;
            fy_[k] = cy - (float)iy;
            fz_[k] = cz - (float)iz;
            base_[k] = (ix << 16) + (iy << 8) + iz;   // x*65536 + y*256 + z
        }

        // ---- phase 2: all gathers issued back-to-back -------------------
        float2 A[4], B[4], C[4], D[4];   // pre-clamped z-pairs (packed path)
        float  s000[4], s001[4], s010[4], s011[4],
               s100[4], s101[4], s110[4], s111[4];
        if (PACKED) {
            #pragma unroll
            for (int k = 0; k < 4; ++k) {
                const int b = base_[k];
                A[k] = grid2[b];            // (x0,y0): c000,c001
                B[k] = grid2[b + 256];      // (x0,y1): c010,c011
                C[k] = grid2[b + 65536];    // (x1,y0): c100,c101
                D[k] = grid2[b + 65792];    // (x1,y1): c110,c111
            }
        } else {
            #pragma unroll
            for (int k = 0; k < 4; ++k) {
                const int b = base_[k];
                s000[k] = grid[b];        s001[k] = grid[b + 1];
                s010[k] = grid[b + 256];  s011[k] = grid[b + 257];
                s100[k] = grid[b + 65536]; s101[k] = grid[b + 65537];
                s110[k] = grid[b + 65792]; s111[k] = grid[b + 65793];
            }
        }

        // ---- phase 3: trilerp + accumulate ------------------------------
        #pragma unroll
        for (int k = 0; k < 4; ++k) {
            float c000, c001, c010, c011, c100, c101, c110, c111;
            if (PACKED) {
                c000 = A[k].x; c001 = A[k].y;
                c010 = B[k].x; c011 = B[k].y;
                c100 = C[k].x; c101 = C[k].y;
                c110 = D[k].x; c111 = D[k].y;
            } else {
                c000 = clamp01(s000[k]); c001 = clamp01(s001[k]);
                c010 = clamp01(s010[k]); c011 = clamp01(s011[k]);
                c100 = clamp01(s100[k]); c101 = clamp01(s101[k]);
                c110 = clamp01(s110[k]); c111 = clamp01(s111[k]);
            }
            const float fz = fz_[k], fy = fy_[k], fx = fx_[k];
            const float c00 = fmaf(fz, c001 - c000, c000);
            const float c01 = fmaf(fz, c011 - c010, c010);
            const float c10 = fmaf(fz, c101 - c100, c100);
            const float c11 = fmaf(fz, c111 - c110, c110);
            const float c0  = fmaf(fy, c01 - c00, c00);
            const float c1  = fmaf(fy, c11 - c10, c10);
            acc += fmaf(fx, c1 - c0, c0);
        }

        gx += dgx4; gy += dgy4; gz += dgz4;
    }

    const float tau = SIGMA_SCALE * dt * acc;
    out[ray] = 1.0f - expf(-tau);
}

extern "C" void kernel_launch(void* const* d_in, const int* in_sizes, int n_in,
                              void* d_out, int out_size, void* d_ws, size_t ws_size,
                              hipStream_t stream) {
    const float* rays = (const float*)d_in[0];   // [65536, 6] fp32
    const float* grid = (const float*)d_in[1];   // [256,256,256] fp32
    float* out = (float*)d_out;                  // [65536] fp32

    (void)in_sizes; (void)n_in; (void)out_size;

    const int threads = 256;                     // 8 wave32s per block
    const int blocks  = N_RAYS / threads;        // 2048 waves total

    if (ws_size >= REPACK_BYTES) {
        float2* grid2 = (float2*)d_ws;
        repack_kernel<<<GRID_ELEMS / 256, 256, 0, stream>>>(grid, grid2);
        nerf_opacity_kernel<true><<<blocks, threads, 0, stream>>>(rays, grid, grid2, out);
    } else {
        nerf_opacity_kernel<false><<<blocks, threads, 0, stream>>>(rays, grid, nullptr, out);
    }
}